// GCNCountry_14422500180194
// MI455X (gfx1250) — compile-verified
//
#include <hip/hip_runtime.h>
#include <hip/hip_bf16.h>

// ---------------------------------------------------------------------------
// GCN forward, algebraically reduced: only row 0 of the network output is
// needed, so  adj @ (x @ W_gc)  collapses to  adj[0,:] @ (x @ W_gc).
// Kernel A fuses the x@W_gc WMMA GEMM tiles with the adj-row-0 weighted
// reduction (h0 is never materialized).  Kernel B runs the tiny MLP tail.
// ---------------------------------------------------------------------------

typedef __attribute__((ext_vector_type(2))) float v2f;
typedef __attribute__((ext_vector_type(8))) float v8f;

#define N_NODES 8192
#define N_FEAT  512
#define N_HID1  1024
#define N_HID2  512
#define LEAKY   0.01f
#define DROP_P  0.3f

__device__ __forceinline__ float leaky_relu(float x) {
    return (x >= 0.0f) ? x : (LEAKY * x);
}

// ---------------------------------------------------------------------------
// Zero the 1024-float accumulator v in workspace.
// ---------------------------------------------------------------------------
__global__ void k_zero(float* __restrict__ v) {
    int i = blockIdx.x * blockDim.x + threadIdx.x;
    if (i < N_HID1) v[i] = 0.0f;
}

// ---------------------------------------------------------------------------
// Fused:  v[n] += sum_m adj[0,m] * (x @ W_gc)[m,n]
// Grid: 64 blocks x 256 threads (8 waves). Each block owns 128 rows of x.
// Each wave computes a 16x16 WMMA tile per N-chunk; the 512x16 W_gc column
// slab is staged in LDS and shared by all 8 waves.
// ---------------------------------------------------------------------------
__global__ void __launch_bounds__(256)
k_fused_gemm_rowred(const float* __restrict__ x,     // [8192,512]
                    const float* __restrict__ adj,   // [8192,8192] (row 0 used)
                    const float* __restrict__ Wgc,   // [512,1024]
                    float* __restrict__ v)           // [1024] accumulator
{
    __shared__ __align__(16) float ws[N_FEAT * 16];  // 32 KB: W_gc[:, n0:n0+16]

    const int tid  = threadIdx.x;
    const int lane = tid & 31;
    const int wv   = tid >> 5;                 // wave id 0..7
    const int hi   = lane >> 4;                // 0 = lanes 0-15, 1 = lanes 16-31
    const int nl   = lane & 15;                // column within tile
    const int klo  = hi * 2;                   // K sub-offset for A/B frags

    const int block_m = blockIdx.x * 128;
    const int wave_m  = block_m + wv * 16;

    // Hoist: adjacency weights for this lane's 8 C-frag rows (M = 8*hi + j).
    const float* awp = adj + wave_m + 8 * hi;  // 32B aligned
    float4 aw0 = *reinterpret_cast<const float4*>(awp);
    float4 aw1 = *reinterpret_cast<const float4*>(awp + 4);
    float aw[8] = {aw0.x, aw0.y, aw0.z, aw0.w, aw1.x, aw1.y, aw1.z, aw1.w};

    // Hoist: this lane's A-matrix row pointer (row M = lane%16 of the tile).
    const float* xrow = x + (size_t)(wave_m + nl) * N_FEAT + klo;

    for (int n0 = 0; n0 < N_HID1; n0 += 16) {
        __syncthreads();  // previous chunk's LDS reads complete
        // Stage W_gc[0:512, n0:n0+16] -> ws (row-major [512][16]), float4s.
        #pragma unroll
        for (int i = 0; i < 8; ++i) {
            int q4  = i * 256 + tid;           // 0..2047
            int row = q4 >> 2;
            int c4  = (q4 & 3) << 2;
            *reinterpret_cast<float4*>(&ws[row * 16 + c4]) =
                *reinterpret_cast<const float4*>(Wgc + (size_t)row * N_HID1 + n0 + c4);
        }
        __syncthreads();

        v8f c = {};
        #pragma unroll 4
        for (int k0 = 0; k0 < N_FEAT; k0 += 4) {
            v2f a = *reinterpret_cast<const v2f*>(xrow + k0);   // A[L%16][k0+klo+{0,1}]
            v2f b;
            b.x = ws[(k0 + klo) * 16 + nl];                     // B[k0+klo  ][L%16]
            b.y = ws[(k0 + klo + 1) * 16 + nl];                 // B[k0+klo+1][L%16]
            c = __builtin_amdgcn_wmma_f32_16x16x4_f32(
                    false, a, false, b, (short)0, c, false, false);
        }

        // Weight C rows by adj[0, row] and fold: lane holds rows M = 8*hi + j.
        float s = 0.0f;
        #pragma unroll
        for (int j = 0; j < 8; ++j) s += aw[j] * c[j];
        // lanes L and L^16 hold the same output column n0 + nl -> fold halves
        s += __shfl_xor(s, 16);
        if (hi == 0) atomicAdd(&v[n0 + nl], s);
    }
}

// ---------------------------------------------------------------------------
// Tail: h1 = leaky(v + b_gc); h2 = leaky(h1 @ W1 + b1); dropout(row 0 mask);
// out  = h2 . W2[:,0] + b2[0].  Single block, 512 threads.
// ---------------------------------------------------------------------------
__global__ void __launch_bounds__(512)
k_tail(const float* __restrict__ v,      // [1024]
       const float* __restrict__ b_gc,   // [1024]
       const float* __restrict__ W1,     // [1024,512]
       const float* __restrict__ b1,     // [512]
       const float* __restrict__ W2,     // [512,1]
       const float* __restrict__ b2,     // [1]
       const float* __restrict__ drop_u, // [8192,512] (row 0 used)
       float* __restrict__ out)          // [1]
{
    __shared__ float h1s[N_HID1];
    __shared__ float red[512];

    const int tid = threadIdx.x;

    for (int j = tid; j < N_HID1; j += 512)
        h1s[j] = leaky_relu(v[j] + b_gc[j]);
    __syncthreads();

    float acc = b1[tid];
    for (int j = 0; j < N_HID1; ++j)
        acc = fmaf(h1s[j], W1[(size_t)j * N_HID2 + tid], acc);
    acc = leaky_relu(acc);

    const float u = drop_u[tid];  // row 0 of the dropout noise
    const float h = (u >= DROP_P) ? acc * (1.0f / (1.0f - DROP_P)) : 0.0f;

    red[tid] = h * W2[tid];       // W2 is [512,1]: column 0 == flat
    __syncthreads();
    #pragma unroll
    for (int s = 256; s > 0; s >>= 1) {
        if (tid < s) red[tid] += red[tid + s];
        __syncthreads();
    }
    if (tid == 0) out[0] = red[0] + b2[0];
}

// ---------------------------------------------------------------------------
extern "C" void kernel_launch(void* const* d_in, const int* in_sizes, int n_in,
                              void* d_out, int out_size, void* d_ws, size_t ws_size,
                              hipStream_t stream) {
    const float* x      = (const float*)d_in[0];
    const float* adj    = (const float*)d_in[1];
    const float* W_gc   = (const float*)d_in[2];
    const float* b_gc   = (const float*)d_in[3];
    const float* W1     = (const float*)d_in[4];
    const float* b1     = (const float*)d_in[5];
    const float* W2     = (const float*)d_in[6];
    const float* b2     = (const float*)d_in[7];
    const float* drop_u = (const float*)d_in[8];

    float* v   = (float*)d_ws;          // 1024 floats of scratch
    float* out = (float*)d_out;

    k_zero<<<4, 256, 0, stream>>>(v);
    k_fused_gemm_rowred<<<N_NODES / 128, 256, 0, stream>>>(x, adj, W_gc, v);
    k_tail<<<1, 512, 0, stream>>>(v, b_gc, W1, b1, W2, b2, drop_u, out);
}